// GATVGAEEncoder_12481174962432
// MI455X (gfx1250) — compile-verified
//
#include <hip/hip_runtime.h>
#include <math.h>

#define N_NODES 50000
#define IN_CH 128
#define HID 32
#define HEADS 4
#define OUT_CH 32
#define N_EDGES 1600000
#define NEG_SLOPE 0.2f

typedef float v2f __attribute__((ext_vector_type(2)));
typedef float v8f __attribute__((ext_vector_type(8)));

static __host__ __device__ inline int cdiv_i(int a, int b) { return (a + b - 1) / b; }

// ---------------------------------------------------------------------------
// WMMA f32 GEMM: C[M,Ncols] = A[M,K] @ B[K,Ncols] + bias[Ncols]
// One wave32 computes one 16x16 tile, K-loop in steps of 4 using
// V_WMMA_F32_16X16X4_F32 (full fp32 precision).
//
// Fragment layout (ISA 7.12.2, 32-bit A 16x4):
//   A: lanes 0-15 hold row M=lane, VGPR0=K0,VGPR1=K1; lanes 16-31: K2,K3
//   B: (symmetric) lanes 0-15 col N=lane, VGPR0=K0,VGPR1=K1; lanes 16-31: K2,K3
//   D: VGPR i -> row (half*8 + i), col = lane%16
// ---------------------------------------------------------------------------
__global__ void gemm_wmma_f32_kernel(const float* __restrict__ A,
                                     const float* __restrict__ B,
                                     const float* __restrict__ bias,
                                     float* __restrict__ C,
                                     int M, int K, int Ncols) {
  int wave = (int)((blockIdx.x * blockDim.x + threadIdx.x) >> 5);
  int lane = (int)(threadIdx.x & 31);
  int tilesN = Ncols >> 4;
  int tilesM = M >> 4;
  if (wave >= tilesM * tilesN) return;
  int tm = wave / tilesN;
  int tn = wave - tm * tilesN;
  int m0 = tm << 4;
  int n0 = tn << 4;
  int half = lane >> 4;   // 0: K{0,1}, 1: K{2,3}
  int l16  = lane & 15;

  const float* arow = A + (size_t)(m0 + l16) * K;
  const float* bcol = B + (size_t)(n0 + l16);

  v8f acc = {};
#if __has_builtin(__builtin_amdgcn_wmma_f32_16x16x4_f32)
  for (int k = 0; k < K; k += 4) {
    v2f a, b;
    a.x = arow[k + half * 2 + 0];
    a.y = arow[k + half * 2 + 1];
    b.x = bcol[(size_t)(k + half * 2 + 0) * Ncols];
    b.y = bcol[(size_t)(k + half * 2 + 1) * Ncols];
    acc = __builtin_amdgcn_wmma_f32_16x16x4_f32(
        /*neg_a=*/false, a, /*neg_b=*/false, b,
        /*c_mod=*/(short)0, acc, /*reuse_a=*/false, /*reuse_b=*/false);
  }
#else
  // Scalar fallback matching the WMMA D layout (compile safety net only).
  for (int i = 0; i < 8; ++i) {
    const float* ar = A + (size_t)(m0 + half * 8 + i) * K;
    float s = 0.0f;
    for (int k = 0; k < K; ++k) s = fmaf(ar[k], bcol[(size_t)k * Ncols], s);
    acc[i] = s;
  }
#endif
  float bb = bias[n0 + l16];
  int row = m0 + half * 8;
#pragma unroll
  for (int i = 0; i < 8; ++i)
    C[(size_t)(row + i) * Ncols + n0 + l16] = acc[i] + bb;
}

// ---------------------------------------------------------------------------
// Small utility kernels
// ---------------------------------------------------------------------------
__global__ void fill_f32_kernel(float* __restrict__ p, float v, int n) {
  int t = blockIdx.x * blockDim.x + threadIdx.x;
  if (t < n) p[t] = v;
}

__global__ void init_bias_kernel(float* __restrict__ out, const float* __restrict__ bias,
                                 int N, int C) {
  int t = blockIdx.x * blockDim.x + threadIdx.x;
  if (t < N * C) out[t] = bias[t % C];
}

__global__ void relu_kernel(float* __restrict__ p, int n) {
  int t = blockIdx.x * blockDim.x + threadIdx.x;
  if (t < n) { float v = p[t]; p[t] = v > 0.0f ? v : 0.0f; }
}

// Pack [Wl | Wr] -> w1[128,256], [bl | br] -> b1c[256]
__global__ void pack_w1_kernel(const float* __restrict__ Wl, const float* __restrict__ bl,
                               const float* __restrict__ Wr, const float* __restrict__ br,
                               float* __restrict__ w1, float* __restrict__ b1c) {
  int t = blockIdx.x * blockDim.x + threadIdx.x;
  if (t < 128 * 256) {
    int k = t >> 8, j = t & 255;
    w1[t] = (j < 128) ? Wl[k * 128 + j] : Wr[k * 128 + (j - 128)];
  }
  if (t < 256) b1c[t] = (t < 128) ? bl[t] : br[t - 128];
}

// Pack [Wl_mu | Wr_mu | Wl_lv | Wr_lv] -> w2[128,128], biases -> b2c[128]
__global__ void pack_w2_kernel(const float* __restrict__ Wlm, const float* __restrict__ blm,
                               const float* __restrict__ Wrm, const float* __restrict__ brm,
                               const float* __restrict__ Wll, const float* __restrict__ bll,
                               const float* __restrict__ Wrl, const float* __restrict__ brl,
                               float* __restrict__ w2, float* __restrict__ b2c) {
  int t = blockIdx.x * blockDim.x + threadIdx.x;
  if (t < 128 * 128) {
    int k = t >> 7, j = t & 127;
    int g = j >> 5, c = j & 31;
    const float* W = (g == 0) ? Wlm : (g == 1) ? Wrm : (g == 2) ? Wll : Wrl;
    w2[t] = W[k * 32 + c];
  }
  if (t < 128) {
    int g = t >> 5, c = t & 31;
    const float* B = (g == 0) ? blm : (g == 1) ? brm : (g == 2) ? bll : brl;
    b2c[t] = B[c];
  }
}

// ---------------------------------------------------------------------------
// Edge kernels. One thread per (edge, head). Edges E..E+N-1 are self loops.
// ---------------------------------------------------------------------------
__global__ void edge_logits_kernel(const int* __restrict__ src, const int* __restrict__ dst,
                                   int E, int N,
                                   const float* __restrict__ xl, const float* __restrict__ xr,
                                   int ld, const float* __restrict__ att, int H, int C,
                                   float* __restrict__ logits, float* __restrict__ mx) {
  int tid = blockIdx.x * blockDim.x + threadIdx.x;
  int tot = (E + N) * H;
  if (tid >= tot) return;
  int e = tid / H, h = tid - e * H;
  int s = (e < E) ? src[e] : (e - E);
  int d = (e < E) ? dst[e] : (e - E);
  const float* pl = xl + (size_t)s * ld + h * C;
  const float* pr = xr + (size_t)d * ld + h * C;
  const float* pa = att + h * C;
  float acc = 0.0f;
  for (int c = 0; c < C; c += 4) {
    float4 a = *(const float4*)(pl + c);
    float4 b = *(const float4*)(pr + c);
    float4 w = *(const float4*)(pa + c);
    float v;
    v = a.x + b.x; v = (v > 0.0f) ? v : NEG_SLOPE * v; acc = fmaf(v, w.x, acc);
    v = a.y + b.y; v = (v > 0.0f) ? v : NEG_SLOPE * v; acc = fmaf(v, w.y, acc);
    v = a.z + b.z; v = (v > 0.0f) ? v : NEG_SLOPE * v; acc = fmaf(v, w.z, acc);
    v = a.w + b.w; v = (v > 0.0f) ? v : NEG_SLOPE * v; acc = fmaf(v, w.w, acc);
  }
  logits[tid] = acc;
  __hip_atomic_fetch_max(&mx[(size_t)d * H + h], acc,
                         __ATOMIC_RELAXED, __HIP_MEMORY_SCOPE_AGENT);
}

__global__ void edge_exp_kernel(const int* __restrict__ dst, int E, int N, int H,
                                float* __restrict__ logits, const float* __restrict__ mx,
                                float* __restrict__ denom) {
  int tid = blockIdx.x * blockDim.x + threadIdx.x;
  int tot = (E + N) * H;
  if (tid >= tot) return;
  int e = tid / H, h = tid - e * H;
  int d = (e < E) ? dst[e] : (e - E);
  float p = __expf(logits[tid] - mx[(size_t)d * H + h]);
  logits[tid] = p;  // overwrite logits with unnormalized p
  atomicAdd(&denom[(size_t)d * H + h], p);
}

__global__ void edge_aggr_kernel(const int* __restrict__ src, const int* __restrict__ dst,
                                 int E, int N,
                                 const float* __restrict__ xl, int ld, int H, int C,
                                 const float* __restrict__ p, const float* __restrict__ denom,
                                 float* __restrict__ out, int ldo) {
  int tid = blockIdx.x * blockDim.x + threadIdx.x;
  int tot = (E + N) * H;
  if (tid >= tot) return;
  int e = tid / H, h = tid - e * H;
  int s = (e < E) ? src[e] : (e - E);
  int d = (e < E) ? dst[e] : (e - E);
  float alpha = p[tid] / denom[(size_t)d * H + h];
  const float* pl = xl + (size_t)s * ld + h * C;
  float* po = out + (size_t)d * ldo + h * C;
  for (int c = 0; c < C; c += 4) {
    float4 a = *(const float4*)(pl + c);
    atomicAdd(&po[c + 0], alpha * a.x);
    atomicAdd(&po[c + 1], alpha * a.y);
    atomicAdd(&po[c + 2], alpha * a.z);
    atomicAdd(&po[c + 3], alpha * a.w);
  }
}

// ---------------------------------------------------------------------------
// Launcher
// ---------------------------------------------------------------------------
extern "C" void kernel_launch(void* const* d_in, const int* in_sizes, int n_in,
                              void* d_out, int out_size, void* d_ws, size_t ws_size,
                              hipStream_t stream) {
  const float* x      = (const float*)d_in[0];
  const int*   ei     = (const int*)  d_in[1];
  const float* Wl1    = (const float*)d_in[2];
  const float* bl1    = (const float*)d_in[3];
  const float* Wr1    = (const float*)d_in[4];
  const float* br1    = (const float*)d_in[5];
  const float* att1   = (const float*)d_in[6];
  const float* b1     = (const float*)d_in[7];
  const float* Wl_mu  = (const float*)d_in[8];
  const float* bl_mu  = (const float*)d_in[9];
  const float* Wr_mu  = (const float*)d_in[10];
  const float* br_mu  = (const float*)d_in[11];
  const float* att_mu = (const float*)d_in[12];
  const float* b_mu   = (const float*)d_in[13];
  const float* Wl_lv  = (const float*)d_in[14];
  const float* bl_lv  = (const float*)d_in[15];
  const float* Wr_lv  = (const float*)d_in[16];
  const float* br_lv  = (const float*)d_in[17];
  const float* att_lv = (const float*)d_in[18];
  const float* b_lv   = (const float*)d_in[19];

  const int N  = N_NODES;
  const int E  = N_EDGES;
  const int ET = E + N;            // with self loops
  const int* src = ei;             // edge_index[0]
  const int* dst = ei + E;         // edge_index[1]

  // Workspace carve-up (floats). Total ~105 MB.
  float* ws     = (float*)d_ws;
  float* w1     = ws;                                  // 128*256
  float* b1c    = w1   + 128 * 256;                    // 256
  float* w2     = b1c  + 256;                          // 128*128
  float* b2c    = w2   + 128 * 128;                    // 128
  float* bufAB  = b2c  + 128;                          // N*256  (xl1|xr1, later layer-2 xfm)
  float* bufH   = bufAB + (size_t)N * 256;             // N*128  (relu'd layer-1 out)
  float* logits = bufH  + (size_t)N * 128;             // ET*HEADS
  float* mx     = logits + (size_t)ET * HEADS;         // N*HEADS
  float* denom  = mx     + (size_t)N * HEADS;          // N*HEADS

  const int B256 = 256;
  float* mu_out = (float*)d_out;           // [N, 32]
  float* lv_out = (float*)d_out + (size_t)N * OUT_CH;  // [N, 32]

  // ---- Layer 1 -------------------------------------------------------------
  pack_w1_kernel<<<cdiv_i(128 * 256, B256), B256, 0, stream>>>(Wl1, bl1, Wr1, br1, w1, b1c);

  {  // [xl1|xr1] = x @ [Wl1|Wr1] + [bl1|br1]  ->  bufAB [N,256]
    int waves = (N / 16) * (256 / 16);
    gemm_wmma_f32_kernel<<<cdiv_i(waves * 32, B256), B256, 0, stream>>>(
        x, w1, b1c, bufAB, N, IN_CH, 256);
  }

  fill_f32_kernel<<<cdiv_i(N * HEADS, B256), B256, 0, stream>>>(mx, -INFINITY, N * HEADS);
  fill_f32_kernel<<<cdiv_i(N * HEADS, B256), B256, 0, stream>>>(denom, 0.0f, N * HEADS);
  init_bias_kernel<<<cdiv_i(N * 128, B256), B256, 0, stream>>>(bufH, b1, N, 128);

  edge_logits_kernel<<<cdiv_i(ET * HEADS, B256), B256, 0, stream>>>(
      src, dst, E, N, bufAB, bufAB + 128, 256, att1, HEADS, HID, logits, mx);
  edge_exp_kernel<<<cdiv_i(ET * HEADS, B256), B256, 0, stream>>>(
      dst, E, N, HEADS, logits, mx, denom);
  edge_aggr_kernel<<<cdiv_i(ET * HEADS, B256), B256, 0, stream>>>(
      src, dst, E, N, bufAB, 256, HEADS, HID, logits, denom, bufH, 128);
  relu_kernel<<<cdiv_i(N * 128, B256), B256, 0, stream>>>(bufH, N * 128);

  // ---- Layer 2 transforms (mu + logvar fused GEMM) -------------------------
  pack_w2_kernel<<<cdiv_i(128 * 128, B256), B256, 0, stream>>>(
      Wl_mu, bl_mu, Wr_mu, br_mu, Wl_lv, bl_lv, Wr_lv, br_lv, w2, b2c);

  {  // bufAB[N,128] = h @ [Wl_mu|Wr_mu|Wl_lv|Wr_lv] + biases
    int waves = (N / 16) * (128 / 16);
    gemm_wmma_f32_kernel<<<cdiv_i(waves * 32, B256), B256, 0, stream>>>(
        bufH, w2, b2c, bufAB, N, 128, 128);
  }

  // ---- mu head (H=1, C=32): xl at col 0, xr at col 32 ----------------------
  fill_f32_kernel<<<cdiv_i(N, B256), B256, 0, stream>>>(mx, -INFINITY, N);
  fill_f32_kernel<<<cdiv_i(N, B256), B256, 0, stream>>>(denom, 0.0f, N);
  init_bias_kernel<<<cdiv_i(N * OUT_CH, B256), B256, 0, stream>>>(mu_out, b_mu, N, OUT_CH);
  edge_logits_kernel<<<cdiv_i(ET, B256), B256, 0, stream>>>(
      src, dst, E, N, bufAB + 0, bufAB + 32, 128, att_mu, 1, OUT_CH, logits, mx);
  edge_exp_kernel<<<cdiv_i(ET, B256), B256, 0, stream>>>(dst, E, N, 1, logits, mx, denom);
  edge_aggr_kernel<<<cdiv_i(ET, B256), B256, 0, stream>>>(
      src, dst, E, N, bufAB + 0, 128, 1, OUT_CH, logits, denom, mu_out, OUT_CH);

  // ---- logvar head (H=1, C=32): xl at col 64, xr at col 96 -----------------
  fill_f32_kernel<<<cdiv_i(N, B256), B256, 0, stream>>>(mx, -INFINITY, N);
  fill_f32_kernel<<<cdiv_i(N, B256), B256, 0, stream>>>(denom, 0.0f, N);
  init_bias_kernel<<<cdiv_i(N * OUT_CH, B256), B256, 0, stream>>>(lv_out, b_lv, N, OUT_CH);
  edge_logits_kernel<<<cdiv_i(ET, B256), B256, 0, stream>>>(
      src, dst, E, N, bufAB + 64, bufAB + 96, 128, att_lv, 1, OUT_CH, logits, mx);
  edge_exp_kernel<<<cdiv_i(ET, B256), B256, 0, stream>>>(dst, E, N, 1, logits, mx, denom);
  edge_aggr_kernel<<<cdiv_i(ET, B256), B256, 0, stream>>>(
      src, dst, E, N, bufAB + 64, 128, 1, OUT_CH, logits, denom, lv_out, OUT_CH);
}